// SSMKernel_90031104459357
// MI455X (gfx1250) — compile-verified
//
#include <hip/hip_runtime.h>
#include <math.h>
#include <stdint.h>

typedef __attribute__((ext_vector_type(2))) float v2f;
typedef __attribute__((ext_vector_type(8))) float v8f;

#define BATCH   8
#define SEQ     2048
#define DIM     1024
#define DSTATE  16
#define CHUNK   64
#define NCHUNK  (SEQ / CHUNK)          // 32
#define TOTCHUNK (BATCH * NCHUNK)      // 256

// ---------------------------------------------------------------------------
// CDNA5 primitives
// ---------------------------------------------------------------------------
__device__ __forceinline__ void lds_fence() {
    asm volatile("s_wait_dscnt 0" ::: "memory");
}
__device__ __forceinline__ void async_wait0() {
    asm volatile("s_wait_asynccnt 0" ::: "memory");
}
// GLOBAL_LOAD_ASYNC_TO_LDS_B128: per-lane 16B global->LDS, tracked by ASYNCcnt
__device__ __forceinline__ void async_copy_b128(void* lds_dst, const void* gsrc) {
    uint32_t l = (uint32_t)(uintptr_t)lds_dst;   // low 32 bits = LDS byte offset
    asm volatile("global_load_async_to_lds_b128 %0, %1, off"
                 :: "v"(l), "v"(gsrc) : "memory");
}

// ---------------------------------------------------------------------------
// wave-level helpers for V_WMMA_F32_16X16X4_F32 (wave32)
// A 16x4:  lanes 0-15 hold M=0..15, vgpr0=K0 vgpr1=K1; lanes 16-31: K2,K3
// B 4x16:  lanes hold N=0..15 twice; vgpr0=K0/K2, vgpr1=K1/K3 (mirror of A)
// C 16x16: vgpr r: lanes 0-15 -> M=r, lanes 16-31 -> M=r+8
// ---------------------------------------------------------------------------
__device__ __forceinline__ void lane_mn(int& m, int& hi) {
    int lane = threadIdx.x & 31;
    m = lane & 15;
    hi = lane >> 4;
}
__device__ __forceinline__ v2f load_a_slice(const float* M, int q) {
    int m, hi; lane_mn(m, hi);
    int k = 4 * q + 2 * hi;
    v2f a; a.x = M[m * 16 + k]; a.y = M[m * 16 + k + 1];
    return a;
}
__device__ __forceinline__ v2f load_b_slice(const float* M, int q) {
    int n, hi; lane_mn(n, hi);
    int k = 4 * q + 2 * hi;
    v2f b; b.x = M[k * 16 + n]; b.y = M[(k + 1) * 16 + n];
    return b;
}
__device__ __forceinline__ void store_c(float* M, v8f c) {
    int n, hi; lane_mn(n, hi);
#pragma unroll
    for (int r = 0; r < 8; ++r) M[(r + 8 * hi) * 16 + n] = c[r];
}
__device__ __forceinline__ v8f wmma4(v2f a, v2f b, v8f c) {
    return __builtin_amdgcn_wmma_f32_16x16x4_f32(false, a, false, b,
                                                 (short)0, c, false, false);
}
// 16x16x16 f32 matmul from row-major LDS operands
__device__ __forceinline__ v8f mm16(const float* A, const float* B, v8f c) {
#pragma unroll
    for (int q = 0; q < 4; ++q)
        c = wmma4(load_a_slice(A, q), load_b_slice(B, q), c);
    return c;
}

// ---------------------------------------------------------------------------
// Stage 1: per 16-position tile: delta, Bu = u@B (WMMA, async-LDS pipelined),
//          A_exp = expm(delta*A) (scaling s=6, degree-8 Horner, 6 squarings)
// grid: BATCH*SEQ/16 blocks of 128 threads (4 waves, wave w owns positions 4w..4w+3)
// ---------------------------------------------------------------------------
__global__ __launch_bounds__(128)
void ssm_stage1(const float* __restrict__ u, const float* __restrict__ gate,
                const float* __restrict__ A_log, const float* __restrict__ Bm,
                const float* __restrict__ dt_w, const float* __restrict__ dt_b,
                float* __restrict__ gAexp, float* __restrict__ gBu) {
    __shared__ float sA[256];
    __shared__ float sDelta[16];
    __shared__ float sPart[4][256];
    __shared__ float sX[4][256];
    __shared__ float sT[4][256];
    __shared__ float sU[4][2][16 * 64];   // per-wave double-buffered u sub-tile

    const int tid  = threadIdx.x;
    const int wave = tid >> 5;
    const int lane = tid & 31;
    const int tile = blockIdx.x;
    const int b    = tile / (SEQ / 16);
    const int s0   = (tile % (SEQ / 16)) * 16;

    // A = -exp(A_log)
    for (int i = tid; i < 256; i += 128) sA[i] = -__expf(A_log[i]);

    // delta = sigmoid(gate . dt_w + dt_b), 4 positions per wave (stream-once -> NT)
    for (int p = wave * 4; p < wave * 4 + 4; ++p) {
        const float* g = gate + ((size_t)b * SEQ + s0 + p) * DIM;
        float acc = 0.f;
        for (int k = lane; k < DIM; k += 32)
            acc += __builtin_nontemporal_load(g + k) * dt_w[k];
#pragma unroll
        for (int off = 16; off > 0; off >>= 1) acc += __shfl_down(acc, off, 32);
        if (lane == 0) sDelta[p] = 1.f / (1.f + __expf(-(acc + dt_b[0])));
    }
    __syncthreads();

    // Bu tile (16x16) = U(16x1024) @ B(1024x16); wave w covers K quarter,
    // async-LDS double-buffered in 16x64 sub-tiles.
    {
        int m, hi; lane_mn(m, hi);
        const float* ub = u + ((size_t)b * SEQ + s0) * DIM;
        float (*ubuf)[16 * 64] = sU[wave];
        const int KQ = wave * 256;

        auto stage = [&](int ci, int bf) {
            const int kb = KQ + ci * 64;
#pragma unroll
            for (int j = 0; j < 8; ++j) {
                int flat = j * 32 + lane;          // 256 chunks of 16B = 4KB
                int row  = flat >> 4;
                int c4   = (flat & 15) * 4;
                async_copy_b128(&ubuf[bf][row * 64 + c4],
                                ub + (size_t)row * DIM + kb + c4);
            }
        };

        stage(0, 0);
        async_wait0();
        v8f c = {};
        int bf = 0;
        for (int ci = 0; ci < 4; ++ci) {
            if (ci < 3) stage(ci + 1, bf ^ 1);     // overlap next stage with math
            const int kb = KQ + ci * 64;
#pragma unroll
            for (int qq = 0; qq < 16; ++qq) {
                int kl = qq * 4 + 2 * hi;
                v2f a;  a.x  = ubuf[bf][m * 64 + kl];
                        a.y  = ubuf[bf][m * 64 + kl + 1];
                int kg = kb + kl;
                v2f bb; bb.x = Bm[kg * DSTATE + m];       // B is 64KB, L2-resident
                        bb.y = Bm[(kg + 1) * DSTATE + m];
                c = wmma4(a, bb, c);
            }
            async_wait0();
            bf ^= 1;
        }
        store_c(sPart[wave], c);
    }
    __syncthreads();
    for (int e = tid; e < 256; e += 128) {
        float t = sPart[0][e] + sPart[1][e] + sPart[2][e] + sPart[3][e];
        int row = e >> 4, n = e & 15;
        gBu[((size_t)b * SEQ + s0 + row) * DSTATE + n] = t;
    }

    // expm: X = delta*A/64; T = I+X/8; T = I+(X*T)/k (k=7..1); square 6x
    float* X = sX[wave];
    float* T = sT[wave];
    int m, hi; lane_mn(m, hi);
    for (int pi = 0; pi < 4; ++pi) {
        const int p = wave * 4 + pi;
        const float dscale = sDelta[p] * (1.0f / 64.0f);
        for (int i = lane; i < 256; i += 32) X[i] = dscale * sA[i];
        lds_fence();
        v2f xa[4];
#pragma unroll
        for (int q = 0; q < 4; ++q) xa[q] = load_a_slice(X, q);
        for (int i = lane; i < 256; i += 32) {
            float idv = ((i >> 4) == (i & 15)) ? 1.f : 0.f;
            T[i] = idv + X[i] * (1.f / 8.f);
        }
        lds_fence();
        for (int k = 7; k >= 1; --k) {
            v8f c = {};
#pragma unroll
            for (int q = 0; q < 4; ++q) c = wmma4(xa[q], load_b_slice(T, q), c);
            lds_fence();
            const float rinv = 1.f / (float)k;
#pragma unroll
            for (int r = 0; r < 8; ++r) {
                int row = r + 8 * hi;
                T[row * 16 + m] = c[r] * rinv + ((row == m) ? 1.f : 0.f);
            }
            lds_fence();
        }
        v8f e = {};
        for (int sq = 0; sq < 6; ++sq) {
            v8f c = {};
            c = mm16(T, T, c);
            e = c;
            if (sq < 5) { lds_fence(); store_c(T, c); lds_fence(); }
        }
        store_c(gAexp + ((size_t)b * SEQ + s0 + p) * 256, e);
    }
}

// ---------------------------------------------------------------------------
// Stage 2a: per chunk (L=64): P_c = M_1*...*M_L (WMMA), v_c = sum b_i*M_{i+1..L}
// A_exp matrices double-buffered into LDS with async copies (prefetch t+1
// while computing t) so the 64-step dependence chain runs at LDS latency.
// grid: 256 blocks of 32 threads
// ---------------------------------------------------------------------------
__global__ __launch_bounds__(32)
void ssm_chunk_reduce(const float* __restrict__ gAexp, const float* __restrict__ gBu,
                      float* __restrict__ gP, float* __restrict__ gv) {
    __shared__ float sP[256];
    __shared__ float sv[16];
    __shared__ float sM[2][256];
    const int lane  = threadIdx.x;
    const int chunk = blockIdx.x;
    const int b     = chunk / NCHUNK;
    const int sbase = (chunk % NCHUNK) * CHUNK;

    for (int i = lane; i < 256; i += 32) sP[i] = ((i >> 4) == (i & 15)) ? 1.f : 0.f;
    if (lane < 16) sv[lane] = 0.f;
    lds_fence();

    auto stageM = [&](int t, int bf) {
        const float* Mg = gAexp + ((size_t)b * SEQ + sbase + t) * 256;
        async_copy_b128(&sM[bf][lane * 8],     Mg + lane * 8);
        async_copy_b128(&sM[bf][lane * 8 + 4], Mg + lane * 8 + 4);
    };

    stageM(0, 0);
    async_wait0();
    for (int t = 0; t < CHUNK; ++t) {
        if (t + 1 < CHUNK) stageM(t + 1, (t + 1) & 1);
        const float* M = sM[t & 1];
        const size_t pos = (size_t)b * SEQ + sbase + t;
        // v' = v*M + Bu_t  (lanes 0..15)
        float vnew = 0.f;
        if (lane < 16) {
            for (int k = 0; k < 16; ++k) vnew += sv[k] * M[k * 16 + lane];
            vnew += gBu[pos * DSTATE + lane];
        }
        // P' = P*M
        v2f pa[4];
#pragma unroll
        for (int q = 0; q < 4; ++q) pa[q] = load_a_slice(sP, q);
        v8f c = {};
#pragma unroll
        for (int q = 0; q < 4; ++q) c = wmma4(pa[q], load_b_slice(M, q), c);
        lds_fence();
        store_c(sP, c);
        if (lane < 16) sv[lane] = vnew;
        lds_fence();
        async_wait0();                     // next-step M resident
    }
    for (int i = lane; i < 256; i += 32) gP[(size_t)chunk * 256 + i] = sP[i];
    if (lane < 16) gv[(size_t)chunk * 16 + lane] = sv[lane];
}

// ---------------------------------------------------------------------------
// Stage 2b: inter-chunk scan per batch: h0[c+1] = h0[c]*P_c + v_c (32 steps)
// grid: BATCH blocks of 32 threads
// ---------------------------------------------------------------------------
__global__ __launch_bounds__(32)
void ssm_chunk_scan(const float* __restrict__ gP, const float* __restrict__ gv,
                    float* __restrict__ gh0) {
    __shared__ float sh[16];
    const int lane = threadIdx.x;
    const int b    = blockIdx.x;
    if (lane < 16) sh[lane] = 0.f;
    lds_fence();
    for (int c = 0; c < NCHUNK; ++c) {
        const size_t ci = (size_t)b * NCHUNK + c;
        if (lane < 16) gh0[ci * 16 + lane] = sh[lane];
        float hn = 0.f;
        if (lane < 16) {
            const float* P = gP + ci * 256;
            for (int k = 0; k < 16; ++k) hn += sh[k] * P[k * 16 + lane];
            hn += gv[ci * 16 + lane];
        }
        lds_fence();
        if (lane < 16) sh[lane] = hn;
        lds_fence();
    }
}

// ---------------------------------------------------------------------------
// Stage 2c: re-walk chunk (async double-buffered M) to produce hs, then
//           y = hs@C^T + D*u as WMMA tiles (NT stores, prefetch on u stream)
// grid: 256 blocks of 32 threads
// ---------------------------------------------------------------------------
__global__ __launch_bounds__(32)
void ssm_expand(const float* __restrict__ gAexp, const float* __restrict__ gBu,
                const float* __restrict__ gh0, const float* __restrict__ u,
                const float* __restrict__ Cm, const float* __restrict__ Dv,
                float* __restrict__ y) {
    __shared__ float sH[CHUNK * 16];
    __shared__ float sh[16];
    __shared__ float sM[2][256];
    const int lane  = threadIdx.x;
    const int chunk = blockIdx.x;
    const int b     = chunk / NCHUNK;
    const int sbase = (chunk % NCHUNK) * CHUNK;

    auto stageM = [&](int t, int bf) {
        const float* Mg = gAexp + ((size_t)b * SEQ + sbase + t) * 256;
        async_copy_b128(&sM[bf][lane * 8],     Mg + lane * 8);
        async_copy_b128(&sM[bf][lane * 8 + 4], Mg + lane * 8 + 4);
    };

    if (lane < 16) sh[lane] = gh0[(size_t)chunk * 16 + lane];
    lds_fence();
    stageM(0, 0);
    async_wait0();
    for (int t = 0; t < CHUNK; ++t) {
        if (t + 1 < CHUNK) stageM(t + 1, (t + 1) & 1);
        const float* M = sM[t & 1];
        const size_t pos = (size_t)b * SEQ + sbase + t;
        float hn = 0.f;
        if (lane < 16) {
            for (int k = 0; k < 16; ++k) hn += sh[k] * M[k * 16 + lane];
            hn += gBu[pos * DSTATE + lane];
        }
        lds_fence();
        if (lane < 16) { sh[lane] = hn; sH[t * 16 + lane] = hn; }
        lds_fence();
        async_wait0();
    }

    // Projection: Y(16pos x 16d) = HS(16x16) @ Ct(16x16) per (pos-tile, d-tile)
    int n, hi; lane_mn(n, hi);
    for (int p = 0; p < 4; ++p) {
        v2f ha[4];
#pragma unroll
        for (int q = 0; q < 4; ++q) {
            int k = 4 * q + 2 * hi;
            v2f a; a.x = sH[(p * 16 + n) * 16 + k];
                   a.y = sH[(p * 16 + n) * 16 + k + 1];
            ha[q] = a;
        }
        const size_t srow0 = (size_t)b * SEQ + sbase + p * 16;
        for (int dt = 0; dt < DIM / 16; ++dt) {
            const int d0 = dt * 16;
            if (dt + 1 < DIM / 16)
                __builtin_prefetch(u + srow0 * DIM + d0 + 16);   // global_prefetch_b8
            v8f c;
#pragma unroll
            for (int r = 0; r < 8; ++r) {
                int row = r + 8 * hi;
                size_t s = srow0 + row;
                c[r] = Dv[d0 + n] * __builtin_nontemporal_load(u + s * DIM + d0 + n);
            }
#pragma unroll
            for (int q = 0; q < 4; ++q) {
                int k = 4 * q + 2 * hi;
                v2f bb; bb.x = Cm[(size_t)(d0 + n) * DSTATE + k];
                        bb.y = Cm[(size_t)(d0 + n) * DSTATE + k + 1];
                c = wmma4(ha[q], bb, c);
            }
#pragma unroll
            for (int r = 0; r < 8; ++r) {
                int row = r + 8 * hi;
                size_t s = srow0 + row;
                __builtin_nontemporal_store(c[r], y + s * DIM + d0 + n);
            }
        }
    }
}

// ---------------------------------------------------------------------------
extern "C" void kernel_launch(void* const* d_in, const int* in_sizes, int n_in,
                              void* d_out, int out_size, void* d_ws, size_t ws_size,
                              hipStream_t stream) {
    const float* u     = (const float*)d_in[0];
    const float* gate  = (const float*)d_in[1];
    const float* A_log = (const float*)d_in[2];
    const float* B_mat = (const float*)d_in[3];
    const float* C_mat = (const float*)d_in[4];
    const float* Dv    = (const float*)d_in[5];
    const float* dt_w  = (const float*)d_in[6];
    const float* dt_b  = (const float*)d_in[7];
    float* y = (float*)d_out;

    float* ws    = (float*)d_ws;
    float* gAexp = ws;                                  // B*S*256      = 4,194,304 f
    float* gBu   = gAexp + (size_t)BATCH * SEQ * 256;   // B*S*16       =   262,144 f
    float* gP    = gBu   + (size_t)BATCH * SEQ * 16;    // 256*256      =    65,536 f
    float* gv    = gP    + (size_t)TOTCHUNK * 256;      // 256*16       =     4,096 f
    float* gh0   = gv    + (size_t)TOTCHUNK * 16;       // 256*16       =     4,096 f

    ssm_stage1<<<dim3(BATCH * SEQ / 16), 128, 0, stream>>>(
        u, gate, A_log, B_mat, dt_w, dt_b, gAexp, gBu);
    ssm_chunk_reduce<<<dim3(TOTCHUNK), 32, 0, stream>>>(gAexp, gBu, gP, gv);
    ssm_chunk_scan<<<dim3(BATCH), 32, 0, stream>>>(gP, gv, gh0);
    ssm_expand<<<dim3(TOTCHUNK), 32, 0, stream>>>(gAexp, gBu, gh0, u, C_mat, Dv, y);
}